// GCNEncoder_6914897347227
// MI455X (gfx1250) — compile-verified
//
#include <hip/hip_runtime.h>
#include <math.h>

typedef __attribute__((ext_vector_type(2))) float v2f;
typedef __attribute__((ext_vector_type(8))) float v8f;

#define BB 16
#define NN 2000
#define HH 128
#define KK 5
#define NROWS (BB * NN)   // 32000
#define LN_EPS 1e-5f

// ---------------------------------------------------------------------------
// Kernel 1: per-row top-5 smallest distances + softmax of (-dist) over them.
// One wave32 per row of dist_matrix (B*N rows, each of length N=2000).
// ---------------------------------------------------------------------------
__device__ __forceinline__ unsigned long long wave_min_u64(unsigned long long key) {
#pragma unroll
  for (int off = 16; off >= 1; off >>= 1) {
    unsigned int lo = (unsigned int)key;
    unsigned int hi = (unsigned int)(key >> 32);
    unsigned int olo = __shfl_xor(lo, off, 32);
    unsigned int ohi = __shfl_xor(hi, off, 32);
    unsigned long long other = (((unsigned long long)ohi) << 32) | (unsigned long long)olo;
    if (other < key) key = other;
  }
  return key;
}

__global__ __launch_bounds__(256) void topk_softmax_kernel(
    const float* __restrict__ dist,   // (B*N, N)
    int* __restrict__ knn_idx,        // (B*N, K)
    float* __restrict__ knn_w)        // (B*N, K)
{
  const int wid = threadIdx.x >> 5;
  const int lane = threadIdx.x & 31;
  const int row = blockIdx.x * 8 + wid;
  if (row >= NROWS) return;

  const float* __restrict__ rp = dist + (size_t)row * NN;

  // local sorted (ascending) top-5 per lane over strided slice
  float ld0 = 3.0e38f, ld1 = 3.0e38f, ld2 = 3.0e38f, ld3 = 3.0e38f, ld4 = 3.0e38f;
  int li0 = 0, li1 = 0, li2 = 0, li3 = 0, li4 = 0;
  for (int j = lane; j < NN; j += 32) {
    float d = rp[j];
    if (d < ld4) {
      ld4 = d; li4 = j;
      if (ld4 < ld3) { float t = ld4; ld4 = ld3; ld3 = t; int ti = li4; li4 = li3; li3 = ti; }
      if (ld3 < ld2) { float t = ld3; ld3 = ld2; ld2 = t; int ti = li3; li3 = li2; li2 = ti; }
      if (ld2 < ld1) { float t = ld2; ld2 = ld1; ld1 = t; int ti = li2; li2 = li1; li1 = ti; }
      if (ld1 < ld0) { float t = ld1; ld1 = ld0; ld0 = t; int ti = li1; li1 = li0; li0 = ti; }
    }
  }

  // 5 rounds of cross-wave min selection on packed (dist_bits, idx) keys
  float sel_d[KK];
  int sel_i[KK];
#pragma unroll
  for (int r = 0; r < KK; r++) {
    unsigned long long mykey =
        (((unsigned long long)__float_as_uint(ld0)) << 32) | (unsigned int)li0;
    unsigned long long win = wave_min_u64(mykey);
    if (mykey == win) {
      // pop my smallest
      ld0 = ld1; li0 = li1;
      ld1 = ld2; li1 = li2;
      ld2 = ld3; li2 = li3;
      ld3 = ld4; li3 = li4;
      ld4 = 3.0e38f; li4 = 0;
    }
    sel_d[r] = __uint_as_float((unsigned int)(win >> 32));
    sel_i[r] = (int)(unsigned int)(win & 0xffffffffu);
  }

  if (lane == 0) {
    // softmax over scores = -d; max score corresponds to smallest dist sel_d[0]
    float e[KK];
    float s = 0.0f;
#pragma unroll
    for (int r = 0; r < KK; r++) {
      e[r] = __expf(sel_d[0] - sel_d[r]);
      s += e[r];
    }
    float inv = 1.0f / s;
    int* ip = knn_idx + (size_t)row * KK;
    float* wp = knn_w + (size_t)row * KK;
#pragma unroll
    for (int r = 0; r < KK; r++) {
      ip[r] = sel_i[r];
      wp[r] = e[r] * inv;
    }
  }
}

// ---------------------------------------------------------------------------
// Kernel 2: one fused GCN layer.
//   agg = sparse(weights) @ h_in          (5-NN gather, weighted sum)
//   delta = relu(agg @ W^T + bias)        (via v_wmma_f32_16x16x4_f32)
//   h_out = layernorm(h_in + delta)
// Block: 256 threads (8 waves). One block = 16 node rows x 128 channels.
// Wave w computes the 16x16 output tile at columns [16w, 16w+16).
// ---------------------------------------------------------------------------
__global__ __launch_bounds__(256) void gcn_layer_kernel(
    const float* __restrict__ h_in,    // (B*N, H)
    const int* __restrict__ knn_idx,   // (B*N, K)
    const float* __restrict__ knn_w,   // (B*N, K)
    const float* __restrict__ W,       // (H, H) row-major: W[o][i]
    const float* __restrict__ bias,    // (H)
    const float* __restrict__ gamma,   // (H)
    const float* __restrict__ beta,    // (H)
    float* __restrict__ h_out)         // (B*N, H)
{
  __shared__ float sW[HH * HH];     // 64 KB: full weight matrix
  __shared__ float sA[16 * HH];     // 8 KB : agg tile (A matrix, row major)
  __shared__ float sS[16 * HH];     // 8 KB : residual + delta (pre-LN)
  __shared__ float red_s[16 * 16];
  __shared__ float red_q[16 * 16];
  __shared__ float row_mu[16];
  __shared__ float row_rstd[16];

  const int tid = threadIdx.x;
  const int tile = blockIdx.x;            // M-tile: rows [tile*16, tile*16+16)
  const int row0 = tile * 16;

  // ---- stage 0: preload W into LDS (coalesced) ----
  for (int i = tid; i < HH * HH; i += 256) sW[i] = W[i];

  // ---- stage 1: build agg tile in LDS ----
  {
    const int i = tid >> 4;               // row in tile 0..15
    const int c0 = (tid & 15) * 8;        // 8 channels per thread
    const int gr = row0 + i;              // global row
    const int b = gr / NN;                // batch
    const int bbase = b * NN;
    float acc0 = 0.f, acc1 = 0.f, acc2 = 0.f, acc3 = 0.f;
    float acc4 = 0.f, acc5 = 0.f, acc6 = 0.f, acc7 = 0.f;
    const int* ip = knn_idx + (size_t)gr * KK;
    const float* wp = knn_w + (size_t)gr * KK;
#pragma unroll
    for (int k = 0; k < KK; k++) {
      const int nb = ip[k];
      const float wk = wp[k];
      const float* src = h_in + (size_t)(bbase + nb) * HH + c0;
      acc0 += wk * src[0]; acc1 += wk * src[1];
      acc2 += wk * src[2]; acc3 += wk * src[3];
      acc4 += wk * src[4]; acc5 += wk * src[5];
      acc6 += wk * src[6]; acc7 += wk * src[7];
    }
    float* dst = &sA[i * HH + c0];
    dst[0] = acc0; dst[1] = acc1; dst[2] = acc2; dst[3] = acc3;
    dst[4] = acc4; dst[5] = acc5; dst[6] = acc6; dst[7] = acc7;
  }
  __syncthreads();

  // ---- stage 2: 16x16 WMMA tile per wave, K = 128 in steps of 4 ----
  const int wave = tid >> 5;              // N-tile index 0..7
  const int lane = tid & 31;
  const int lane16 = lane & 15;
  const int hi = lane >> 4;               // 0: K pair {0,1}, 1: K pair {2,3}
  const int nb = wave * 16;
  const int ncol = nb + lane16;           // this lane's output channel

  v8f acc = {};
#pragma unroll 4
  for (int kb = 0; kb < HH; kb += 4) {
    const int ka = kb + (hi << 1);
    // A fragment: A[m][k], m = lane16, k = ka, ka+1
    v2f afrag = *(const v2f*)&sA[lane16 * HH + ka];
    // B fragment: B[k][n] = W^T[k][n] = W[n][k], n = ncol
    v2f bfrag = *(const v2f*)&sW[ncol * HH + ka];
    acc = __builtin_amdgcn_wmma_f32_16x16x4_f32(
        /*neg_a=*/false, afrag, /*neg_b=*/false, bfrag,
        /*c_mod=*/(short)0, acc, /*reuse_a=*/false, /*reuse_b=*/false);
  }

  // ---- stage 3: bias + relu + residual -> sS ----
  {
    const float bn = bias[ncol];
#pragma unroll
    for (int r = 0; r < 8; r++) {
      const int m = r + (hi << 3);        // D layout: lanes16-31 hold rows 8..15
      const int gr = row0 + m;
      float delta = acc[r] + bn;
      delta = fmaxf(delta, 0.0f);
      const float res = h_in[(size_t)gr * HH + ncol];
      sS[m * HH + ncol] = res + delta;
    }
  }
  __syncthreads();

  // ---- stage 4: per-row LayerNorm over 128 channels, write h_out ----
  {
    const int i = tid >> 4;               // row 0..15
    const int sub = tid & 15;             // 16 partials per row
    const int c0 = sub * 8;
    float s1 = 0.f, s2 = 0.f;
#pragma unroll
    for (int j = 0; j < 8; j++) {
      float v = sS[i * HH + c0 + j];
      s1 += v;
      s2 += v * v;
    }
    red_s[i * 16 + sub] = s1;
    red_q[i * 16 + sub] = s2;
    __syncthreads();
    if (sub == 0) {
      float a = 0.f, q = 0.f;
#pragma unroll
      for (int t = 0; t < 16; t++) { a += red_s[i * 16 + t]; q += red_q[i * 16 + t]; }
      const float mu = a * (1.0f / HH);
      const float var = q * (1.0f / HH) - mu * mu;
      row_mu[i] = mu;
      row_rstd[i] = rsqrtf(var + LN_EPS);
    }
    __syncthreads();
    const float mu = row_mu[i];
    const float rstd = row_rstd[i];
    float* op = h_out + (size_t)(row0 + i) * HH + c0;
#pragma unroll
    for (int j = 0; j < 8; j++) {
      const int c = c0 + j;
      op[j] = (sS[i * HH + c] - mu) * rstd * gamma[c] + beta[c];
    }
  }
}

// ---------------------------------------------------------------------------
// Host launcher
// ---------------------------------------------------------------------------
extern "C" void kernel_launch(void* const* d_in, const int* in_sizes, int n_in,
                              void* d_out, int out_size, void* d_ws, size_t ws_size,
                              hipStream_t stream) {
  const float* node_emb = (const float*)d_in[0];   // (B,N,H)
  const float* dist     = (const float*)d_in[1];   // (B,N,N)
  const float* Ws       = (const float*)d_in[2];   // (L,H,H)
  const float* bs       = (const float*)d_in[3];   // (L,H)
  const float* gammas   = (const float*)d_in[4];   // (L,H)
  const float* betas    = (const float*)d_in[5];   // (L,H)
  float* out = (float*)d_out;

  char* ws = (char*)d_ws;
  int*   knn_idx = (int*)ws;                                 // 32000*5*4   = 640 KB
  float* knn_w   = (float*)(ws + (size_t)NROWS * KK * 4);    // 640 KB
  float* h_mid   = (float*)(ws + 2 * (size_t)NROWS * KK * 4);// 16 MB ping buffer

  // Stage A: kNN + softmax -> compact (idx, w)
  topk_softmax_kernel<<<NROWS / 8, 256, 0, stream>>>(dist, knn_idx, knn_w);

  // Stage B: 3 fused layers, ping-pong d_out <-> h_mid, final lands in d_out.
  gcn_layer_kernel<<<NROWS / 16, 256, 0, stream>>>(
      node_emb, knn_idx, knn_w, Ws + 0 * HH * HH, bs + 0 * HH,
      gammas + 0 * HH, betas + 0 * HH, out);
  gcn_layer_kernel<<<NROWS / 16, 256, 0, stream>>>(
      out, knn_idx, knn_w, Ws + 1 * HH * HH, bs + 1 * HH,
      gammas + 1 * HH, betas + 1 * HH, h_mid);
  gcn_layer_kernel<<<NROWS / 16, 256, 0, stream>>>(
      h_mid, knn_idx, knn_w, Ws + 2 * HH * HH, bs + 2 * HH,
      gammas + 2 * HH, betas + 2 * HH, out);
}